// LightGCN_12257836662826
// MI455X (gfx1250) — compile-verified
//
#include <hip/hip_runtime.h>
#include <stdint.h>

#define EMB 64
#define TILE 256        // edges per tile == blockDim of scatter kernel

// ---------------------------------------------------------------------------
// CDNA5 async global->LDS helpers (VGLOBAL async ops, tracked by ASYNCcnt).
// LDS address: generic shared pointers are {SHARED_BASE, lds_offset}; the low
// 32 bits are the wave-relative LDS byte offset the instruction expects.
// ---------------------------------------------------------------------------
__device__ __forceinline__ uint32_t lds_off(const void* p) {
    return (uint32_t)(uint64_t)p;
}
__device__ __forceinline__ void async_load_b64(uint32_t lds, const void* gaddr) {
    asm volatile("global_load_async_to_lds_b64 %0, %1, off"
                 :: "v"(lds), "v"(gaddr) : "memory");
}
__device__ __forceinline__ void async_load_b32(uint32_t lds, const void* gaddr) {
    asm volatile("global_load_async_to_lds_b32 %0, %1, off"
                 :: "v"(lds), "v"(gaddr) : "memory");
}

// ---------------------------------------------------------------------------
// init: cur = acc = concat(user_emb, item_emb)   (float4 vectorized)
// Explicit branch so each path does exactly ONE b128 load feeding BOTH stores
// (round-1 disasm showed the select-of-pointers form re-loaded per store).
// ---------------------------------------------------------------------------
__global__ void lgcn_init(const float* __restrict__ user_w,
                          const float* __restrict__ item_w,
                          float* __restrict__ cur, float* __restrict__ acc,
                          int userVec4, int totalVec4) {
    int i = blockIdx.x * blockDim.x + threadIdx.x;
    if (i >= totalVec4) return;
    if (i < userVec4) {
        float4 v = ((const float4*)user_w)[i];
        ((float4*)cur)[i] = v;
        ((float4*)acc)[i] = v;
    } else {
        float4 v = ((const float4*)item_w)[i - userVec4];
        ((float4*)cur)[i] = v;
        ((float4*)acc)[i] = v;
    }
}

// ---------------------------------------------------------------------------
// accum: acc = (acc + nxt) * s    (s = 1 for layers 0..1, 0.25 folded into last)
// ---------------------------------------------------------------------------
__global__ void lgcn_accum(float* __restrict__ acc, const float* __restrict__ nxt,
                           float s, int totalVec4) {
    int i = blockIdx.x * blockDim.x + threadIdx.x;
    if (i >= totalVec4) return;
    float4 a = ((const float4*)acc)[i];
    float4 b = ((const float4*)nxt)[i];
    a.x = (a.x + b.x) * s;
    a.y = (a.y + b.y) * s;
    a.z = (a.z + b.z) * s;
    a.w = (a.w + b.w) * s;
    ((float4*)acc)[i] = a;
}

// ---------------------------------------------------------------------------
// scatter: nxt[row[e]] += cur[col[e]] * w[e]   for all edges.
// 16 lanes per edge (each lane owns a float4 chunk of the 64-dim row):
//  - gather is one contiguous 256B row per half-wave (global_load_b128);
//    cur (76.8 MB) is L2-resident so repeated gathers hit L2
//  - scatter-add is 4 global_atomic_add_f32 per lane (no return -> STOREcnt);
//    nxt (76.8 MB) is also L2-resident, so the RMWs stay in L2
// Edge tiles (row,col,w) are double-buffered in LDS via async global->LDS
// loads; the next tile is prefetched while the current one is consumed
// (s_wait_asynccnt 3 keeps the prefetch in flight; async loads complete
//  in order, so <=3 outstanding guarantees the current tile has landed).
// ---------------------------------------------------------------------------
__global__ __launch_bounds__(TILE) void lgcn_scatter(
    const float* __restrict__ cur, float* __restrict__ nxt,
    const long long* __restrict__ row, const long long* __restrict__ col,
    const float* __restrict__ w, int E, int numTiles) {

    __shared__ long long s_row[2][TILE];
    __shared__ long long s_col[2][TILE];
    __shared__ float     s_w[2][TILE];

    const int tid = threadIdx.x;
    const int g = tid >> 4;          // edge slot within iteration (0..15)
    const int l = tid & 15;          // float4 chunk within the 64-dim row
    const int stride = gridDim.x;
    const float4* __restrict__ cur4 = (const float4*)cur;

    int t = blockIdx.x;
    if (t >= numTiles) return;

    // Stage first tile into buffer 0 (3 async ops per wave).
    {
        int e = t * TILE + tid; if (e >= E) e = E - 1;
        async_load_b64(lds_off(&s_row[0][tid]), row + e);
        async_load_b64(lds_off(&s_col[0][tid]), col + e);
        async_load_b32(lds_off(&s_w[0][tid]),   w + e);
    }

    int buf = 0;
    for (; t < numTiles; t += stride, buf ^= 1) {
        const int nt = t + stride;
        if (nt < numTiles) {
            int e = nt * TILE + tid; if (e >= E) e = E - 1;
            async_load_b64(lds_off(&s_row[buf ^ 1][tid]), row + e);
            async_load_b64(lds_off(&s_col[buf ^ 1][tid]), col + e);
            async_load_b32(lds_off(&s_w[buf ^ 1][tid]),   w + e);
            asm volatile("s_wait_asynccnt 3" ::: "memory");   // current tile done
        } else {
            asm volatile("s_wait_asynccnt 0" ::: "memory");
        }
        __syncthreads();   // all waves' staged data visible to the workgroup

        const int base = t * TILE;
        int n = E - base; if (n > TILE) n = TILE;

        #pragma unroll 4
        for (int it = 0; it < TILE / 16; ++it) {
            const int e = (it << 4) + g;
            if (e < n) {
                const int   r  = (int)s_row[buf][e];
                const int   c  = (int)s_col[buf][e];
                const float wt = s_w[buf][e];
                // Hide latency of the next random gather (global_prefetch_b8).
                const int e2 = e + 16;
                if (e2 < n) {
                    const int c2 = (int)s_col[buf][e2];
                    __builtin_prefetch(&cur4[(size_t)c2 * 16 + l], 0, 3);
                }
                const float4 v = cur4[(size_t)c * 16 + l];        // b128 gather
                float* dst = nxt + (size_t)r * EMB + (l << 2);
                __hip_atomic_fetch_add(dst + 0, v.x * wt, __ATOMIC_RELAXED, __HIP_MEMORY_SCOPE_AGENT);
                __hip_atomic_fetch_add(dst + 1, v.y * wt, __ATOMIC_RELAXED, __HIP_MEMORY_SCOPE_AGENT);
                __hip_atomic_fetch_add(dst + 2, v.z * wt, __ATOMIC_RELAXED, __HIP_MEMORY_SCOPE_AGENT);
                __hip_atomic_fetch_add(dst + 3, v.w * wt, __ATOMIC_RELAXED, __HIP_MEMORY_SCOPE_AGENT);
            }
        }
        __syncthreads();   // everyone done reading buf before it is restaged
    }
}

// ---------------------------------------------------------------------------
// Host-side launch sequence (graph-capture safe: only kernels + memsetAsync).
// ---------------------------------------------------------------------------
extern "C" void kernel_launch(void* const* d_in, const int* in_sizes, int n_in,
                              void* d_out, int out_size, void* d_ws, size_t ws_size,
                              hipStream_t stream) {
    const float*     user_w = (const float*)d_in[0];
    const float*     item_w = (const float*)d_in[1];
    const long long* eidx   = (const long long*)d_in[2];   // int64 [2, E]
    const float*     ew     = (const float*)d_in[3];

    const int n_user  = in_sizes[0] / EMB;
    const int n_item  = in_sizes[1] / EMB;
    const int n_nodes = n_user + n_item;
    const int E       = in_sizes[3];
    const long long* row = eidx;        // edge_index[0]
    const long long* col = eidx + E;    // edge_index[1]

    float* acc = (float*)d_out;
    float* cur = (float*)d_ws;                             // [n_nodes * 64]
    float* nxt = cur + (size_t)n_nodes * EMB;              // [n_nodes * 64]
    const size_t embBytes = (size_t)n_nodes * EMB * sizeof(float);

    const int vec4 = n_nodes * (EMB / 4);
    const int ewBlocks = (vec4 + 255) / 256;

    lgcn_init<<<ewBlocks, 256, 0, stream>>>(user_w, item_w, cur, acc,
                                            n_user * (EMB / 4), vec4);

    const int numTiles = (E + TILE - 1) / TILE;
    const int nBlocks  = numTiles < 4096 ? numTiles : 4096;  // persistent blocks

    for (int layer = 0; layer < 3; ++layer) {
        hipMemsetAsync(nxt, 0, embBytes, stream);
        lgcn_scatter<<<nBlocks, TILE, 0, stream>>>(cur, nxt, row, col, ew, E, numTiles);
        const float s = (layer == 2) ? 0.25f : 1.0f;         // fold /4 into last pass
        lgcn_accum<<<ewBlocks, 256, 0, stream>>>(acc, nxt, s, vec4);
        float* tmp = cur; cur = nxt; nxt = tmp;
    }
}